// DataNet0_39049842655640
// MI455X (gfx1250) — compile-verified
//
#include <hip/hip_runtime.h>
#include <math.h>

// ---------------------------------------------------------------------------
// Fourier ptychography on MI455X (gfx1250).
// FFTs are dense DFT matmuls on V_WMMA_F32_16X16X4_F32 (precision is free:
// workload is serial-chain latency bound, ~0.3 TFLOP total).
// fftshift/ifftshift folded into the DFT matrices:
//   Fs[k][n] = exp(-2*pi*i*k*n/N) * (-1)^n        (fftshift(fft2(x)) = Fs x Fs^T)
//   Bs[m][n] = exp(+2*pi*i*m*n/N) * (-1)^m / N    (ifft2(ifftshift(P)) = Bs P Bs^T)
// T*L = 147 sequential updates run in ONE persistent kernel, one workgroup
// (1024 threads / 32 wave32) per batch image. Each wave register-blocks an
// M-pair of 16x16 tiles sharing B fragments (8 WMMA per K-chunk, 6 loads).
// LDS: 256 KB ping-pong intermediates + 64 KB Y buffer filled by
// GLOBAL_LOAD_ASYNC_TO_LDS_B128 (ASYNCcnt), = 320 KB (max per workgroup).
// ---------------------------------------------------------------------------

#define BATCH 32
#define LNUM  49
#define TITER 3
#define NSZ   256
#define HLR   128

typedef __attribute__((ext_vector_type(2))) float v2f;
typedef __attribute__((ext_vector_type(8))) float v8f;

#define WMMAF32(a, b, c) \
  __builtin_amdgcn_wmma_f32_16x16x4_f32(false, (a), false, (b), (short)0, (c), false, false)

// ---- WMMA fragment helpers (wave32, 16x16x4 f32) ---------------------------
// A (16x4 f32): lanes 0-15 -> M=lane, K={k0,k0+1}; lanes 16-31 -> K={k0+2,k0+3}
__device__ __forceinline__ v2f load_a(const float* A, int lda, int m0, int k0, int lane) {
  const int m = m0 + (lane & 15);
  const int k = k0 + ((lane >> 4) << 1);
  return *(const v2f*)(A + m * lda + k);
}
// B (4x16 f32), row-major B[k][n]
__device__ __forceinline__ v2f load_b(const float* B, int ldb, int k0, int n0, int lane) {
  const int n = n0 + (lane & 15);
  const int k = k0 + ((lane >> 4) << 1);
  v2f r;
  r.x = B[(k    ) * ldb + n];
  r.y = B[(k + 1) * ldb + n];
  return r;
}
// B accessed transposed: element (k,n) = Bt[n][k] (contiguous in k -> b64 load)
__device__ __forceinline__ v2f load_bt(const float* Bt, int ldb, int k0, int n0, int lane) {
  const int n = n0 + (lane & 15);
  const int k = k0 + ((lane >> 4) << 1);
  return *(const v2f*)(Bt + n * ldb + k);
}
// C/D (16x16 f32): VGPR r: lanes 0-15 -> M=r, lanes 16-31 -> M=r+8, N=lane&15
__device__ __forceinline__ void store_c(float* C, int ldc, int m0, int n0, int lane, v8f c) {
  const int n  = n0 + (lane & 15);
  const int mb = m0 + ((lane >> 4) << 3);
#pragma unroll
  for (int r = 0; r < 8; ++r) C[(mb + r) * ldc + n] = c[r];
}

// Complex MMA, M-pair register blocked: two 16x16 accumulators share B frags.
// f32 WMMA NEG only supports C-negate (ISA 7.12) -> negate Ai in VALU.
__device__ __forceinline__ void cmma2(v2f ar0, v2f ai0, v2f ar1, v2f ai1,
                                      v2f br, v2f bi,
                                      v8f& c0r, v8f& c0i, v8f& c1r, v8f& c1i) {
  v2f n0; n0.x = -ai0.x; n0.y = -ai0.y;
  v2f n1; n1.x = -ai1.x; n1.y = -ai1.y;
  c0r = WMMAF32(ar0, br, c0r);
  c0r = WMMAF32(n0,  bi, c0r);
  c0i = WMMAF32(ar0, bi, c0i);
  c0i = WMMAF32(ai0, br, c0i);
  c1r = WMMAF32(ar1, br, c1r);
  c1r = WMMAF32(n1,  bi, c1r);
  c1i = WMMAF32(ar1, bi, c1i);
  c1i = WMMAF32(ai1, br, c1i);
}

// ---- CDNA5 async memory -> LDS (ASYNCcnt) ----------------------------------
// Copy one 16384-float Y slice into the LDS Y buffer (64 KB), 1024 threads,
// 4 x b128 per thread. LDS dest offset = static LDS size + 256 KB (the Y
// buffer sits after the 4 intermediate planes in dynamic LDS).
__device__ __forceinline__ void prefetch_y_async(const float* src, int tid) {
  const unsigned base = (unsigned)__builtin_amdgcn_groupstaticsize() + 262144u;
#pragma unroll
  for (int j = 0; j < 4; ++j) {
    const int e = j * 4096 + tid * 4;
    const float* g = src + e;
    const unsigned lds = base + (unsigned)e * 4u;
    asm volatile("global_load_async_to_lds_b128 %0, %1, off" :: "v"(lds), "v"(g) : "memory");
  }
}
__device__ __forceinline__ void wait_async0() {
  asm volatile("s_wait_asynccnt 0x0" ::: "memory");
}

// ---------------------------------------------------------------------------
// Kernel 0: build shifted DFT matrices in workspace.
// ---------------------------------------------------------------------------
__global__ void k_build_mats(float* Fs256r, float* Fs256i, float* Bs256r, float* Bs256i,
                             float* Fs128r, float* Fs128i, float* Bs128r, float* Bs128i) {
  const float TWO_PI = 6.283185307179586f;
  for (int idx = blockIdx.x * blockDim.x + threadIdx.x; idx < NSZ * NSZ;
       idx += gridDim.x * blockDim.x) {
    {
      const int k = idx >> 8, n = idx & 255;
      const int p = (k * n) & 255;
      const float a = TWO_PI * (float)p / 256.0f;
      const float c = cosf(a), s = sinf(a);
      const float sn = (n & 1) ? -1.0f : 1.0f;
      const float sk = (k & 1) ? -1.0f : 1.0f;
      Fs256r[idx] = c * sn;
      Fs256i[idx] = -s * sn;
      Bs256r[idx] = c * sk * (1.0f / 256.0f);
      Bs256i[idx] = s * sk * (1.0f / 256.0f);
    }
    if (idx < HLR * HLR) {
      const int k = idx >> 7, n = idx & 127;
      const int p = (k * n) & 127;
      const float a = TWO_PI * (float)p / 128.0f;
      const float c = cosf(a), s = sinf(a);
      const float sn = (n & 1) ? -1.0f : 1.0f;
      const float sk = (k & 1) ? -1.0f : 1.0f;
      Fs128r[idx] = c * sn;
      Fs128i[idx] = -s * sn;
      Bs128r[idx] = c * sk * (1.0f / 128.0f);
      Bs128i[idx] = s * sk * (1.0f / 128.0f);
    }
  }
}

// ---------------------------------------------------------------------------
// Batched complex GEMM: C = A*B or A*B^T, S x S. One wave computes a 32x16
// strip (M-pair of tiles) sharing B fragments. Stride 0 => shared matrix.
// ---------------------------------------------------------------------------
__global__ __launch_bounds__(256)
void k_cgemm(const float* __restrict__ Ar, const float* __restrict__ Ai, long sa,
             const float* __restrict__ Br, const float* __restrict__ Bi, long sb,
             float* __restrict__ Cr, float* __restrict__ Ci, long sc,
             int S, int btrans) {
  const int b    = blockIdx.y;
  const int lane = threadIdx.x & 31;
  const int wave = threadIdx.x >> 5;
  const int tilesN = S >> 4;
  const int task   = blockIdx.x * 8 + wave;
  if (task >= (tilesN >> 1) * tilesN) return;
  const int m0 = (task / tilesN) << 5;
  const int n0 = (task % tilesN) << 4;
  const float* ar_ = Ar + (long)b * sa;
  const float* ai_ = Ai + (long)b * sa;
  const float* br_ = Br + (long)b * sb;
  const float* bi_ = Bi + (long)b * sb;
  v8f c0r = {}, c0i = {}, c1r = {}, c1i = {};
  for (int k0 = 0; k0 < S; k0 += 4) {
    v2f ar0 = load_a(ar_, S, m0,      k0, lane);
    v2f ai0 = load_a(ai_, S, m0,      k0, lane);
    v2f ar1 = load_a(ar_, S, m0 + 16, k0, lane);
    v2f ai1 = load_a(ai_, S, m0 + 16, k0, lane);
    v2f xbr, xbi;
    if (btrans) { xbr = load_bt(br_, S, k0, n0, lane); xbi = load_bt(bi_, S, k0, n0, lane); }
    else        { xbr = load_b (br_, S, k0, n0, lane); xbi = load_b (bi_, S, k0, n0, lane); }
    cmma2(ar0, ai0, ar1, ai1, xbr, xbi, c0r, c0i, c1r, c1i);
  }
  float* cr_ = Cr + (long)b * sc;
  float* ci_ = Ci + (long)b * sc;
  store_c(cr_, S, m0,      n0, lane, c0r);
  store_c(ci_, S, m0,      n0, lane, c0i);
  store_c(cr_, S, m0 + 16, n0, lane, c1r);
  store_c(ci_, S, m0 + 16, n0, lane, c1i);
}

// ---------------------------------------------------------------------------
// Persistent iterate kernel: one workgroup per batch image, 32 wave32.
// Each wave owns one 32x16 strip (M-pair x 1 N-tile) => 32 tasks = 64 tiles.
// ---------------------------------------------------------------------------
__global__ __launch_bounds__(1024)
void k_fp_iterate(float* __restrict__ OFTr, float* __restrict__ OFTi,
                  const float* __restrict__ F128r, const float* __restrict__ F128i,
                  const float* __restrict__ B128r, const float* __restrict__ B128i,
                  const float* __restrict__ CTF, const float* __restrict__ Y,
                  const int* __restrict__ Masks) {
  extern __shared__ float smem[];
  float* s0r = smem;                 // 16384 floats each
  float* s0i = smem + 16384;
  float* s1r = smem + 32768;
  float* s1i = smem + 49152;
  const float* ylds = smem + 65536;  // 16384 floats: async Y buffer

  const int b    = blockIdx.x;
  const int tid  = threadIdx.x;
  const int lane = tid & 31;
  const int wave = tid >> 5;                 // 0..31
  const int m0   = (wave >> 3) << 5;         // 0,32,64,96
  const int n0   = (wave & 7) << 4;          // 0..112
  const int n    = n0 + (lane & 15);

  float* oftr = OFTr + (long)b * (NSZ * NSZ);
  float* ofti = OFTi + (long)b * (NSZ * NSZ);
  const float* Yb = Y + (long)b * LNUM * (HLR * HLR);

  // Prologue: start async copy of Y[l=0] into LDS.
  prefetch_y_async(Yb, tid);

  for (int t = 0; t < TITER; ++t) {
    for (int l = 0; l < LNUM; ++l) {
      const int ix = Masks[2 * l] - 1;
      const int iy = Masks[2 * l + 1] - 1;

      // ---- Stage 1: s0 = Bs128 @ (patch .* CTF)
      {
        v8f c0r = {}, c0i = {}, c1r = {}, c1i = {};
        for (int k0 = 0; k0 < HLR; k0 += 4) {
          v2f ar0 = load_a(B128r, HLR, m0,      k0, lane);
          v2f ai0 = load_a(B128i, HLR, m0,      k0, lane);
          v2f ar1 = load_a(B128r, HLR, m0 + 16, k0, lane);
          v2f ai1 = load_a(B128i, HLR, m0 + 16, k0, lane);
          const int kk = k0 + ((lane >> 4) << 1);
          const float w0 = CTF[kk * HLR + n];
          const float w1 = CTF[(kk + 1) * HLR + n];
          v2f br, bi;
          br.x = oftr[(ix + kk    ) * NSZ + iy + n] * w0;
          br.y = oftr[(ix + kk + 1) * NSZ + iy + n] * w1;
          bi.x = ofti[(ix + kk    ) * NSZ + iy + n] * w0;
          bi.y = ofti[(ix + kk + 1) * NSZ + iy + n] * w1;
          cmma2(ar0, ai0, ar1, ai1, br, bi, c0r, c0i, c1r, c1i);
        }
        store_c(s0r, HLR, m0,      n0, lane, c0r);
        store_c(s0i, HLR, m0,      n0, lane, c0i);
        store_c(s0r, HLR, m0 + 16, n0, lane, c1r);
        store_c(s0i, HLR, m0 + 16, n0, lane, c1i);
      }
      wait_async0();        // Y[l] fully in LDS (each wave waits its own xfers)
      __syncthreads();

      // ---- Stage 2: Img_low = s0 @ Bs128^T ; amplitude replacement -> s1
      {
        v8f c0r = {}, c0i = {}, c1r = {}, c1i = {};
        for (int k0 = 0; k0 < HLR; k0 += 4) {
          v2f ar0 = load_a(s0r, HLR, m0,      k0, lane);
          v2f ai0 = load_a(s0i, HLR, m0,      k0, lane);
          v2f ar1 = load_a(s0r, HLR, m0 + 16, k0, lane);
          v2f ai1 = load_a(s0i, HLR, m0 + 16, k0, lane);
          v2f br  = load_bt(B128r, HLR, k0, n0, lane);
          v2f bi  = load_bt(B128i, HLR, k0, n0, lane);
          cmma2(ar0, ai0, ar1, ai1, br, bi, c0r, c0i, c1r, c1i);
        }
#pragma unroll
        for (int h = 0; h < 2; ++h) {
          const v8f cre = h ? c1r : c0r;
          const v8f cim = h ? c1i : c0i;
          const int mb = m0 + 16 * h + ((lane >> 4) << 3);
#pragma unroll
          for (int r = 0; r < 8; ++r) {
            const int row = mb + r;
            const float re = cre[r], im = cim[r];
            const float mag = sqrtf(re * re + im * im);
            const float amp = sqrtf(ylds[row * HLR + n]);
            const float inv = amp / mag;                    // selected away if mag==0
            s1r[row * HLR + n] = (mag > 0.0f) ? re * inv : amp;  // angle(0)=0
            s1i[row * HLR + n] = (mag > 0.0f) ? im * inv : 0.0f;
          }
        }
      }
      __syncthreads();      // all yamp reads + s1 writes complete

      // Start async copy of the NEXT step's Y slice (overlaps stages 3,4,1).
      {
        const int ln = (l + 1 < LNUM) ? l + 1 : 0;
        prefetch_y_async(Yb + (long)ln * (HLR * HLR), tid);
      }

      // ---- Stage 3: s0 = Fs128 @ s1
      {
        v8f c0r = {}, c0i = {}, c1r = {}, c1i = {};
        for (int k0 = 0; k0 < HLR; k0 += 4) {
          v2f ar0 = load_a(F128r, HLR, m0,      k0, lane);
          v2f ai0 = load_a(F128i, HLR, m0,      k0, lane);
          v2f ar1 = load_a(F128r, HLR, m0 + 16, k0, lane);
          v2f ai1 = load_a(F128i, HLR, m0 + 16, k0, lane);
          v2f br  = load_b(s1r, HLR, k0, n0, lane);
          v2f bi  = load_b(s1i, HLR, k0, n0, lane);
          cmma2(ar0, ai0, ar1, ai1, br, bi, c0r, c0i, c1r, c1i);
        }
        store_c(s0r, HLR, m0,      n0, lane, c0r);
        store_c(s0i, HLR, m0,      n0, lane, c0i);
        store_c(s0r, HLR, m0 + 16, n0, lane, c1r);
        store_c(s0i, HLR, m0 + 16, n0, lane, c1i);
      }
      __syncthreads();

      // ---- Stage 4: Object_part = (s0 @ Fs128^T) .* CTF ; blend into OFT
      {
        v8f c0r = {}, c0i = {}, c1r = {}, c1i = {};
        for (int k0 = 0; k0 < HLR; k0 += 4) {
          v2f ar0 = load_a(s0r, HLR, m0,      k0, lane);
          v2f ai0 = load_a(s0i, HLR, m0,      k0, lane);
          v2f ar1 = load_a(s0r, HLR, m0 + 16, k0, lane);
          v2f ai1 = load_a(s0i, HLR, m0 + 16, k0, lane);
          v2f br  = load_bt(F128r, HLR, k0, n0, lane);
          v2f bi  = load_bt(F128i, HLR, k0, n0, lane);
          cmma2(ar0, ai0, ar1, ai1, br, bi, c0r, c0i, c1r, c1i);
        }
#pragma unroll
        for (int h = 0; h < 2; ++h) {
          const v8f cre = h ? c1r : c0r;
          const v8f cim = h ? c1i : c0i;
          const int mb = m0 + 16 * h + ((lane >> 4) << 3);
#pragma unroll
          for (int r = 0; r < 8; ++r) {
            const int row = mb + r;
            const float ctf = CTF[row * HLR + n];
            const long  g   = (long)(ix + row) * NSZ + iy + n;
            const float pr = oftr[g], pi = ofti[g];
            oftr[g] = (1.0f - ctf) * pr + cre[r] * ctf;
            ofti[g] = (1.0f - ctf) * pi + cim[r] * ctf;
          }
        }
      }
      __threadfence();
      __syncthreads();
    }
  }
}

// ---------------------------------------------------------------------------
// Final stage: im = T @ Bs256^T with fused abs/clip, angle, complex outputs.
// d_out layout: [ im_ra (B*N*N) | im_rp (B*N*N) | im_rc interleaved (2*B*N*N) ]
// ---------------------------------------------------------------------------
__global__ __launch_bounds__(256)
void k_final(const float* __restrict__ Tr, const float* __restrict__ Ti,
             const float* __restrict__ Bsr, const float* __restrict__ Bsi,
             float* __restrict__ out) {
  const int b    = blockIdx.y;
  const int lane = threadIdx.x & 31;
  const int wave = threadIdx.x >> 5;
  const int task = blockIdx.x * 8 + wave;      // 128 tasks
  const int m0 = (task >> 4) << 5;
  const int n0 = (task & 15) << 4;
  const float* ar_ = Tr + (long)b * (NSZ * NSZ);
  const float* ai_ = Ti + (long)b * (NSZ * NSZ);
  v8f c0r = {}, c0i = {}, c1r = {}, c1i = {};
  for (int k0 = 0; k0 < NSZ; k0 += 4) {
    v2f ar0 = load_a(ar_, NSZ, m0,      k0, lane);
    v2f ai0 = load_a(ai_, NSZ, m0,      k0, lane);
    v2f ar1 = load_a(ar_, NSZ, m0 + 16, k0, lane);
    v2f ai1 = load_a(ai_, NSZ, m0 + 16, k0, lane);
    v2f br  = load_bt(Bsr, NSZ, k0, n0, lane);
    v2f bi  = load_bt(Bsi, NSZ, k0, n0, lane);
    cmma2(ar0, ai0, ar1, ai1, br, bi, c0r, c0i, c1r, c1i);
  }
  const long NN = (long)NSZ * NSZ;
  float* out_ra = out;
  float* out_rp = out + BATCH * NN;
  float* out_rc = out + 2L * BATCH * NN;
  const int n = n0 + (lane & 15);
#pragma unroll
  for (int h = 0; h < 2; ++h) {
    const v8f cre = h ? c1r : c0r;
    const v8f cim = h ? c1i : c0i;
    const int mb = m0 + 16 * h + ((lane >> 4) << 3);
#pragma unroll
    for (int r = 0; r < 8; ++r) {
      const long idx = (long)b * NN + (long)(mb + r) * NSZ + n;
      const float re = cre[r], im = cim[r];
      const float mag = sqrtf(re * re + im * im);
      out_ra[idx] = fminf(mag, 100.0f);
      out_rp[idx] = atan2f(im, re);
      out_rc[2 * idx]     = re;
      out_rc[2 * idx + 1] = im;
    }
  }
}

// ---------------------------------------------------------------------------
extern "C" void kernel_launch(void* const* d_in, const int* in_sizes, int n_in,
                              void* d_out, int out_size, void* d_ws, size_t ws_size,
                              hipStream_t stream) {
  const float* Img_real = (const float*)d_in[0];
  const float* Img_imag = (const float*)d_in[1];
  const float* Y        = (const float*)d_in[2];
  const int*   Masks    = (const int*)d_in[3];
  const float* CTF      = (const float*)d_in[4];
  // d_in[5..7] = n1, n2, T (device scalars); shapes are compile-time constants here.

  float* ws = (float*)d_ws;
  const long NN256 = (long)NSZ * NSZ;     // 65536
  const long NN128 = (long)HLR * HLR;     // 16384
  float* Fs256r = ws;                 float* Fs256i = Fs256r + NN256;
  float* Bs256r = Fs256i + NN256;     float* Bs256i = Bs256r + NN256;
  float* Fs128r = Bs256i + NN256;     float* Fs128i = Fs128r + NN128;
  float* Bs128r = Fs128i + NN128;     float* Bs128i = Bs128r + NN128;
  float* OFTr   = Bs128i + NN128;     float* OFTi   = OFTr + BATCH * NN256;
  float* Tr     = OFTi + BATCH * NN256;
  float* Ti     = Tr + BATCH * NN256;
  const size_t need = (size_t)((Ti + BATCH * NN256) - ws) * sizeof(float);
  if (ws_size < need) return;

  // 0) shifted DFT matrices
  k_build_mats<<<64, 256, 0, stream>>>(Fs256r, Fs256i, Bs256r, Bs256i,
                                       Fs128r, Fs128i, Bs128r, Bs128i);

  // 1) Object_FT = Fs256 @ Img @ Fs256^T
  k_cgemm<<<dim3(16, BATCH), 256, 0, stream>>>(
      Fs256r, Fs256i, 0L, Img_real, Img_imag, NN256, Tr, Ti, NN256, NSZ, 0);
  k_cgemm<<<dim3(16, BATCH), 256, 0, stream>>>(
      Tr, Ti, NN256, Fs256r, Fs256i, 0L, OFTr, OFTi, NN256, NSZ, 1);

  // 2) T*L sequential FP updates, one persistent workgroup per batch image.
  //    Dynamic LDS: 4 intermediate planes (256 KB) + async Y buffer (64 KB).
  k_fp_iterate<<<dim3(BATCH), 1024, 5 * HLR * HLR * sizeof(float), stream>>>(
      OFTr, OFTi, Fs128r, Fs128i, Bs128r, Bs128i, CTF, Y, Masks);

  // 3) im_rc = Bs256 @ Object_FT @ Bs256^T, fused abs/clip + angle + complex out
  k_cgemm<<<dim3(16, BATCH), 256, 0, stream>>>(
      Bs256r, Bs256i, 0L, OFTr, OFTi, NN256, Tr, Ti, NN256, NSZ, 0);
  k_final<<<dim3(16, BATCH), 256, 0, stream>>>(Tr, Ti, Bs256r, Bs256i, (float*)d_out);
}